// Sentence_Rep_2052994367502
// MI455X (gfx1250) — compile-verified
//
#include <hip/hip_runtime.h>

typedef __attribute__((ext_vector_type(2))) float v2f;
typedef __attribute__((ext_vector_type(8))) float v8f;

#define BATCH  2048
#define SEQ    200
#define DIMN   300
#define DTILES 19              // ceil(300/16)
#define NTHREADS (DTILES * 32) // 19 waves * wave32 = 608
#define MASKWORDS 7            // ceil(200 bits / 32)

// One block per sentence b. Wave w reduces columns [16w, 16w+16) over all 200
// rows with V_WMMA_F32_16X16X4_F32 (A = all-ones 16x4 -> every D row = column
// sums). Row-nonzero flags are gathered branchlessly: wave-uniform ballot
// nibbles are OR-accumulated into a 200-bit mask distributed across lanes
// (lane j holds rows 32j..32j+31), merged across waves with 7 ds_or's at the
// end. Numerator masking is algebraically redundant (zero rows add zero), so
// this is a single streaming pass: 494 MB @ 23.3 TB/s ~= 21 us floor.
__global__ __launch_bounds__(NTHREADS) void sentence_rep_kernel(
    const float* __restrict__ w, float* __restrict__ out)
{
    __shared__ unsigned rowmask[MASKWORDS];
    __shared__ int cnt;

    const int tid  = threadIdx.x;
    const int lane = tid & 31;
    const int wv   = tid >> 5;
    const int b    = blockIdx.x;

    if (tid < MASKWORDS) rowmask[tid] = 0u;
    if (tid == 0)        cnt = 0;
    __syncthreads();

    const int  c    = lane & 15;        // column within tile == N slot (L%16)
    const int  half = lane >> 4;        // 0: rows s,s+1   1: rows s+2,s+3
    const int  col  = wv * 16 + c;
    const bool inb  = col < DIMN;
    const int  colc = inb ? col : (DIMN - 1);   // clamp address, zero the value

    const float* base = w + (size_t)b * SEQ * DIMN + colc;

    v8f acc = {};              // C/D accumulator: 8 VGPRs, all rows identical
    v2f aones; aones.x = 1.0f; aones.y = 1.0f;  // A = 16x4 ones

    unsigned mybits = 0u;      // lane j accumulates flags for rows 32j..32j+31

    for (int i = 0; i < SEQ / 4; ++i) {
        const int s  = i * 4;
        const int r0 = s + half * 2;
        float x0 = base[(size_t)r0 * DIMN];             // row r0
        float x1 = base[(size_t)r0 * DIMN + DIMN];      // row r0+1
        if (!inb) { x0 = 0.0f; x1 = 0.0f; }

        // Row nonzero detection from the data we already loaded (uniform SGPRs).
        unsigned m0 = __builtin_amdgcn_ballot_w32(x0 != 0.0f);
        unsigned m1 = __builtin_amdgcn_ballot_w32(x1 != 0.0f);
        unsigned nib = ((m0 & 0xFFFFu) ? 1u : 0u)   // row s   (lanes 0-15, x0)
                     | ((m1 & 0xFFFFu) ? 2u : 0u)   // row s+1 (lanes 0-15, x1)
                     | ((m0 >> 16)     ? 4u : 0u)   // row s+2 (lanes 16-31, x0)
                     | ((m1 >> 16)     ? 8u : 0u);  // row s+3 (lanes 16-31, x1)
        // Branchless scatter of the nibble to the owning lane/bit position.
        mybits |= (lane == (i >> 3)) ? (nib << ((i & 7) * 4)) : 0u;

        v2f bv; bv.x = x0; bv.y = x1;    // B = 4x16 tile of this sentence
        // D = ones(16x4) * B(4x16) + C  ->  every D row = 4-row column sums
        acc = __builtin_amdgcn_wmma_f32_16x16x4_f32(
            /*neg_a=*/false, aones, /*neg_b=*/false, bv,
            /*c_mod=*/(short)0, acc, /*reuse_a=*/false, /*reuse_b=*/false);
    }

    // Merge this wave's 200-bit row mask into the block-wide mask.
    if (lane < MASKWORDS) atomicOr(&rowmask[lane], mybits);
    __syncthreads();
    if (tid < MASKWORDS) atomicAdd(&cnt, __popc(rowmask[tid]));
    __syncthreads();

    const float inv = 1.0f / (float)cnt;     // >=1 guaranteed by reference
    // acc[0] at lane L holds D[M=(half?8:0), N=L&15]; all D rows identical, so
    // lanes 0-15 carry the full set of 16 column sums for this tile.
    if (half == 0 && inb)
        out[(size_t)b * DIMN + col] = acc[0] * inv;
}

extern "C" void kernel_launch(void* const* d_in, const int* in_sizes, int n_in,
                              void* d_out, int out_size, void* d_ws, size_t ws_size,
                              hipStream_t stream) {
    (void)in_sizes; (void)n_in; (void)out_size; (void)d_ws; (void)ws_size;
    const float* w  = (const float*)d_in[0];
    float* out      = (float*)d_out;
    sentence_rep_kernel<<<BATCH, NTHREADS, 0, stream>>>(w, out);
}